// HighGCN_84327387890023
// MI455X (gfx1250) — compile-verified
//
#include <hip/hip_runtime.h>

// HighGCN forward for gfx1250 (MI455X), fp32 end-to-end via V_WMMA_F32_16X16X4_F32.
// Memory-bound (200MB X read dominates; 51MB feature mats live in 192MB L2), so
// fp32 WMMA gives reference precision at negligible cost vs bf16.
// W is staged in LDS K-pair-interleaved so each WMMA B-fragment is one aligned
// 64-bit DS load; staging uses GLOBAL_LOAD_ASYNC_TO_LDS (ASYNCcnt path).
// B fragments are batch-loaded per k-step so DS latency overlaps the WMMA chain.

#define N_NODES   100000
#define N_EDGES   1600000
#define INIT_DIM  500
#define D_MODEL   128
#define CLASS_NUM 129
#define NEG_SLOPE 0.01f
#define LDW       144   // padded col count; 2*LDW % 64 == 32 -> bank-clean b64 frags

typedef __attribute__((ext_vector_type(2))) float v2f;
typedef __attribute__((ext_vector_type(8))) float v8f;

#if __has_builtin(__builtin_amdgcn_global_load_async_to_lds_b32)
#define HAVE_ASYNC_LDS 1
typedef __attribute__((address_space(1))) int* gptr_i;
typedef __attribute__((address_space(3))) int* lptr_i;
#else
#define HAVE_ASYNC_LDS 0
#endif

// ---------------------------------------------------------------- utilities
__global__ void fill_zero_kernel(float* __restrict__ p, long long n) {
    long long i = blockIdx.x * (long long)blockDim.x + threadIdx.x;
    long long stride = (long long)gridDim.x * blockDim.x;
    for (; i < n; i += stride) p[i] = 0.0f;
}

__global__ void degree_kernel(const int* __restrict__ src, const int* __restrict__ dst,
                              float* __restrict__ deg_out, float* __restrict__ deg_in,
                              int n_edges) {
    int e = blockIdx.x * blockDim.x + threadIdx.x;
    if (e >= n_edges) return;
    unsafeAtomicAdd(&deg_out[src[e]], 1.0f);   // global_atomic_add_f32
    unsafeAtomicAdd(&deg_in[dst[e]], 1.0f);
}

__global__ void rsqrt_clip_kernel(float* __restrict__ p, int n) {
    int i = blockIdx.x * blockDim.x + threadIdx.x;
    if (i < n) p[i] = rsqrtf(fmaxf(p[i], 1.0f));
}

// ---------------------------------------------------------------- SpMM scatter
// agg[dst] += x[src] * rs_out[src]; one wave per edge, 4 floats per lane.
__global__ void aggregate_kernel(const int* __restrict__ src, const int* __restrict__ dst,
                                 const float* __restrict__ x, const float* __restrict__ rs_out,
                                 float* __restrict__ agg, int n_edges) {
    long long gid = blockIdx.x * (long long)blockDim.x + threadIdx.x;
    if (gid >= (long long)n_edges * 32) return;
    int e = (int)(gid >> 5);
    int c = (int)(gid & 31) * 4;
    int s = src[e], d = dst[e];
    float sc = rs_out[s];
    const float4 v = *(const float4*)(x + (size_t)s * D_MODEL + c);
    float* p = agg + (size_t)d * D_MODEL + c;
    unsafeAtomicAdd(p + 0, v.x * sc);
    unsafeAtomicAdd(p + 1, v.y * sc);
    unsafeAtomicAdd(p + 2, v.z * sc);
    unsafeAtomicAdd(p + 3, v.w * sc);
}

// ---------------------------------------------------------------- WMMA GEMM
// out[n_rows x ncol] = act( (A * rs_row) @ W + bias ).
// W staged in LDS K-pair-interleaved: element (k,c) -> dword (k>>1)*2*LDW + 2c + (k&1),
// so the f32 16x16x4 B-fragment (K=k+kb, K=k+kb+1 at col) is one aligned 64-bit load.
// Fragment layout per ISA 7.12.2: lane<16 -> K0/K1, lane>=16 -> K2/K3.
template <int NT>
__global__ void gemm_wmma_kernel(const float* __restrict__ A, int K,
                                 const float* __restrict__ W, int ncol,
                                 const float* __restrict__ bias,
                                 const float* __restrict__ rs,
                                 float* __restrict__ out, int n_rows,
                                 int KC, int leaky) {
    extern __shared__ float sW[];                    // [KC/2][2*LDW] pair-interleaved
    const int tid  = threadIdx.x;
    const int wave = tid >> 5;
    const int lane = tid & 31;
    const int hi   = lane >> 4;                      // 0 or 1
    const int lo   = lane & 15;
    const int kb   = hi * 2;                         // K sub-offset within a 4-step
    const int row_base = blockIdx.x * 128 + wave * 16;

    int row = row_base + lo;
    if (row >= n_rows) row = n_rows - 1;             // clamp: EXEC stays all-ones
    const float scale = rs ? rs[row] : 1.0f;
    const float* Arow = A + (size_t)row * K;

    // zero LDS once (pads beyond ncol stay zero across chunks)
    for (int idx = tid; idx < KC * LDW; idx += blockDim.x) sW[idx] = 0.0f;
    __syncthreads();

    v8f acc[NT] = {};

    for (int kc = 0; kc < K; kc += KC) {
        const int ck = (K - kc < KC) ? (K - kc) : KC;
        // stage W chunk into interleaved LDS (per-element async scatter, ASYNCcnt)
#if HAVE_ASYNC_LDS
        for (int idx = tid; idx < ck * ncol; idx += blockDim.x) {
            int k = idx / ncol, c = idx - k * ncol;
            int ld = (k >> 1) * (2 * LDW) + (c << 1) + (k & 1);
            __builtin_amdgcn_global_load_async_to_lds_b32(
                (gptr_i)(W + (size_t)(kc + k) * ncol + c),
                (lptr_i)(sW + ld), 0, 0);
        }
        asm volatile("s_wait_asynccnt 0x0" ::: "memory");
#else
        for (int idx = tid; idx < ck * ncol; idx += blockDim.x) {
            int k = idx / ncol, c = idx - k * ncol;
            sW[(k >> 1) * (2 * LDW) + (c << 1) + (k & 1)] = W[(size_t)(kc + k) * ncol + c];
        }
#endif
        __syncthreads();

        for (int k = 0; k < ck; k += 4) {
            v2f a = *(const v2f*)(Arow + kc + k + kb);   // aligned: K, kc, k, kb all even
            // batch-load all NT B fragments so DS latency overlaps the WMMA chain
            const float* bp = sW + ((k + kb) >> 1) * (2 * LDW) + (lo << 1);
            v2f bf[NT];
#pragma unroll
            for (int t = 0; t < NT; ++t) bf[t] = *(const v2f*)(bp + (t << 5));
            a.x *= scale;
            a.y *= scale;
#pragma unroll
            for (int t = 0; t < NT; ++t) {
                acc[t] = __builtin_amdgcn_wmma_f32_16x16x4_f32(
                    false, a, false, bf[t], (short)0, acc[t], false, false);
            }
        }
        __syncthreads();
    }

    // epilogue: VGPR r of C holds row (r + 8*hi), col = t*16 + lo
    const int out_row0 = row_base + hi * 8;
#pragma unroll
    for (int t = 0; t < NT; ++t) {
        const int col = t * 16 + lo;
        if (col >= ncol) continue;
        const float bb = bias ? bias[col] : 0.0f;
#pragma unroll
        for (int r = 0; r < 8; ++r) {
            const int orow = out_row0 + r;
            if (orow < n_rows) {
                float v = acc[t][r] + bb;
                if (leaky) v = (v >= 0.0f) ? v : NEG_SLOPE * v;
                out[(size_t)orow * ncol + col] = v;
            }
        }
    }
}

// ---------------------------------------------------------------- gate
// t = dot(e_row, Wg) + bg;  prev_row = e_row*t + prev_row*(1-t)   (in place)
__global__ void gate_kernel(const float* __restrict__ e, const float* __restrict__ Wg,
                            const float* __restrict__ bg, float* __restrict__ prev_inout,
                            int n_rows) {
    int row  = blockIdx.x * (blockDim.x >> 5) + (threadIdx.x >> 5);
    int lane = threadIdx.x & 31;
    if (row >= n_rows) return;
    const float4 ev = *(const float4*)(e + (size_t)row * D_MODEL + lane * 4);
    const float4 wv = *(const float4*)(Wg + lane * 4);
    float local = ev.x * wv.x + ev.y * wv.y + ev.z * wv.z + ev.w * wv.w;
#pragma unroll
    for (int m = 16; m >= 1; m >>= 1) local += __shfl_xor(local, m, 32);
    const float t  = local + bg[0];
    const float u  = 1.0f - t;
    float4* pp = (float4*)(prev_inout + (size_t)row * D_MODEL + lane * 4);
    float4 pv = *pp;
    float4 o;
    o.x = ev.x * t + pv.x * u;
    o.y = ev.y * t + pv.y * u;
    o.z = ev.z * t + pv.z * u;
    o.w = ev.w * t + pv.w * u;
    *pp = o;
}

// ---------------------------------------------------------------- launch
extern "C" void kernel_launch(void* const* d_in, const int* in_sizes, int n_in,
                              void* d_out, int out_size, void* d_ws, size_t ws_size,
                              hipStream_t stream) {
    const int*   src  = (const int*)d_in[0];
    const int*   dst  = (const int*)d_in[1];
    const float* X    = (const float*)d_in[2];
    const float* W_fc = (const float*)d_in[3];
    const float* b_fc = (const float*)d_in[4];
    const float* W_g1 = (const float*)d_in[5];
    const float* W_g2 = (const float*)d_in[6];
    const float* b_g2 = (const float*)d_in[7];
    const float* W_g3 = (const float*)d_in[8];
    const float* b_g3 = (const float*)d_in[9];
    const float* W_t1 = (const float*)d_in[10];
    const float* b_t1 = (const float*)d_in[11];
    const float* W_t2 = (const float*)d_in[12];
    const float* b_t2 = (const float*)d_in[13];
    float* out = (float*)d_out;

    const int N = N_NODES, E = N_EDGES;

    // workspace: 3 rotating [N,128] feature buffers + rs_out/rs_in
    float* B0     = (float*)d_ws;
    float* B1     = B0 + (size_t)N * D_MODEL;
    float* B2     = B1 + (size_t)N * D_MODEL;
    float* rs     = B2 + (size_t)N * D_MODEL;
    float* rs_out = rs;
    float* rs_in  = rs + N;

    const dim3 blk(256);
    const int gemm_blocks = (N + 127) / 128;
    const int agg_blocks  = (int)(((long long)E * 32 + 255) / 256);
    const int gate_blocks = (N + 7) / 8;

    // degrees -> rsqrt(clip(deg,1))
    fill_zero_kernel<<<512, blk, 0, stream>>>(rs, 2LL * N);
    degree_kernel<<<(E + 255) / 256, blk, 0, stream>>>(src, dst, rs_out, rs_in, E);
    rsqrt_clip_kernel<<<(2 * N + 255) / 256, blk, 0, stream>>>(rs, 2 * N);

    // FC: B0 = X @ W_fc + b_fc        (K=500, chunk 100 -> 100*144*4 = 57.6KB LDS)
    gemm_wmma_kernel<8><<<gemm_blocks, blk, 100 * LDW * 4, stream>>>(
        X, INIT_DIM, W_fc, D_MODEL, b_fc, nullptr, B0, N, 100, 0);

    // conv1: e0 = leaky((D^-1/2 A D^-1/2 B0) @ W_g1)
    fill_zero_kernel<<<2048, blk, 0, stream>>>(B1, (long long)N * D_MODEL);
    aggregate_kernel<<<agg_blocks, blk, 0, stream>>>(src, dst, B0, rs_out, B1, E);
    gemm_wmma_kernel<8><<<gemm_blocks, blk, 64 * LDW * 4, stream>>>(
        B1, D_MODEL, W_g1, D_MODEL, nullptr, rs_in, B2, N, 64, 1);
    // gate1: B0 = e0*t1 + g_feat*(1-t1)
    gate_kernel<<<gate_blocks, blk, 0, stream>>>(B2, W_t1, b_t1, B0, N);

    // conv2 (+bias) and gate2
    fill_zero_kernel<<<2048, blk, 0, stream>>>(B1, (long long)N * D_MODEL);
    aggregate_kernel<<<agg_blocks, blk, 0, stream>>>(src, dst, B0, rs_out, B1, E);
    gemm_wmma_kernel<8><<<gemm_blocks, blk, 64 * LDW * 4, stream>>>(
        B1, D_MODEL, W_g2, D_MODEL, b_g2, rs_in, B2, N, 64, 1);
    gate_kernel<<<gate_blocks, blk, 0, stream>>>(B2, W_t2, b_t2, B0, N);

    // conv3 -> out [N,129]  (NT=9 col tiles; pads in LDS stay zero)
    fill_zero_kernel<<<2048, blk, 0, stream>>>(B1, (long long)N * D_MODEL);
    aggregate_kernel<<<agg_blocks, blk, 0, stream>>>(src, dst, B0, rs_out, B1, E);
    gemm_wmma_kernel<9><<<gemm_blocks, blk, 64 * LDW * 4, stream>>>(
        B1, D_MODEL, W_g3, CLASS_NUM, b_g3, rs_in, out, N, 64, 1);

    (void)in_sizes; (void)n_in; (void)out_size; (void)ws_size;
}